// Warp_58153857188504
// MI455X (gfx1250) — compile-verified
//
#include <hip/hip_runtime.h>

// Bilinear warp (gather) for img[8,64,256,448] + flow[8,2,256,448], f32.
// Memory-bound (~480 MB @ 23.3 TB/s ~ 21 us). Pipeline per 16-pixel tile:
//   1) coalesced gathers (lane = pixel) of the 4 bilinear corners, per channel
//   2) LDS transpose into WMMA B[4x16] layout (conflict-free b64 ops)
//   3) V_WMMA_F32_16X16X4_F32 (exact f32, K=4 corners, block-diagonal A)
//   4) LDS transpose of D tiles -> coalesced channels-first stores

typedef float v2f __attribute__((ext_vector_type(2)));
typedef float v8f __attribute__((ext_vector_type(8)));

#define Bn 8
#define Cn 64
#define Hn 256
#define Wn 448
#define HW (Hn * Wn)
#define CHW (Cn * HW)
#define TILES_X (Wn / 16)            // 28
#define NTILES (Bn * Hn * TILES_X)   // 57344
#define WAVES_PER_BLOCK 8
#define BSTR 68                      // B-stage row stride (floats): 68 % 64 == 4 -> conflict-free
#define OSTR 19                      // out-transpose row stride: conflict-free both phases
#define LDS_FLOATS 1216              // max(16*BSTR=1088, 64*OSTR=1216)

__global__ __launch_bounds__(256) void warp_bilinear_wmma(
    const float* __restrict__ img, const float* __restrict__ flo,
    float* __restrict__ out)
{
    __shared__ float lds[WAVES_PER_BLOCK][LDS_FLOATS];

    const int lane  = threadIdx.x & 31;
    // force wave-uniform scalars into SGPRs -> saddr-form loads, scalar bases
    const int wave  = __builtin_amdgcn_readfirstlane(threadIdx.x >> 5);
    const int tile  = blockIdx.x * WAVES_PER_BLOCK + wave;

    const int wt = tile % TILES_X;
    const int hb = tile / TILES_X;
    const int h  = hb % Hn;
    const int b  = hb / Hn;
    const int w0 = wt * 16;

    const int  px = lane & 15;       // this lane's pixel (gather) / channel (B,N)
    const bool hi = lane >= 16;
    const int  kh = hi ? 2 : 0;      // corner-pair: lo lanes K=0,1  hi lanes K=2,3
    const int  w  = w0 + px;

    // ---- per-pixel flow, corners, weights (lanes 16-31 mirror 0-15) ----
    const float* fp = flo + (size_t)b * 2 * HW + (size_t)h * Wn + w;
    float x = (float)w + fp[0];
    float y = (float)h + fp[HW];

    int x0 = (int)x;                 // trunc toward zero == tf.cast(int32)
    int y0 = (int)y;
    int x1 = x0 + 1, y1 = y0 + 1;
    x0 = min(max(x0, 0), Wn - 1);
    x1 = min(max(x1, 0), Wn - 1);
    y0 = min(max(y0, 0), Hn - 1);
    y1 = min(max(y1, 0), Hn - 1);

    float dx1 = (float)x1 - x, dx0 = x - (float)x0;
    float dy1 = (float)y1 - y, dy0 = y - (float)y0;
    float wa = dx1 * dy1;            // K=0: (y0,x0)
    float wb = dx1 * dy0;            // K=1: (y1,x0)
    float wc = dx0 * dy1;            // K=2: (y0,x1)
    float wd = dx0 * dy0;            // K=3: (y1,x1)

    // A 16x4 layout: lanes0-15 hold K=0/1, lanes16-31 hold K=2/3 of row m=lane%16
    const float wX = hi ? wc : wa;
    const float wY = hi ? wd : wb;

    // per-lane gather byte offsets within one channel plane, reused for all 64 channels
    const int offXb = 4 * (hi ? (y0 * Wn + x1) : (y0 * Wn + x0));  // K=0 / K=2 corner
    const int offYb = 4 * (hi ? (y1 * Wn + x1) : (y1 * Wn + x0));  // K=1 / K=3 corner

    const float* gbase = img + (size_t)b * CHW;       // SGPR base
    float* buf = &lds[wave][0];                       // wave-private scratch
    const int sCol = px * 4 + kh;                     // column in B-stage row

    v8f acc[4] = {};

#pragma unroll
    for (int g = 0; g < 4; ++g) {
        // ---- stage 16 channels: coalesced gathers (lane = pixel) ----
#pragma unroll
        for (int cl = 0; cl < 16; ++cl) {
            const char* plane = (const char*)(gbase + (size_t)(g * 16 + cl) * HW);
            v2f pk;
            pk.x = *(const float*)(plane + offXb);    // corner K / K   (A or C)
            pk.y = *(const float*)(plane + offYb);    // corner K+1     (B or D)
            *(v2f*)&buf[cl * BSTR + sCol] = pk;       // ds_store_b64, conflict-free
        }
        asm volatile("s_wait_dscnt 0x0" ::: "memory");

        // ---- consume: 16 block-diagonal WMMAs fill the 16 pixel rows ----
#pragma unroll
        for (int p = 0; p < 16; ++p) {
            v2f a;
            a.x = (px == p) ? wX : 0.0f;
            a.y = (px == p) ? wY : 0.0f;
            // B[4x16]: lane reads its channel row, pixel p, its corner pair
            v2f bm = *(const v2f*)&buf[px * BSTR + p * 4 + kh];  // ds_load_b64
            acc[g] = __builtin_amdgcn_wmma_f32_16x16x4_f32(
                false, a, false, bm, (short)0, acc[g], false, false);
        }
        asm volatile("s_wait_dscnt 0x0" ::: "memory"); // buffer reuse (WAR) guard
    }

    // ---- transpose D tiles through LDS: buf[c][pixel] ----
    // C/D layout: VGPR r holds M=r (lanes 0-15) and M=r+8 (lanes 16-31), N=lane%16
    const int mbase = hi ? 8 : 0;
#pragma unroll
    for (int g = 0; g < 4; ++g)
#pragma unroll
        for (int r = 0; r < 8; ++r)
            buf[(g * 16 + px) * OSTR + (r + mbase)] = acc[g][r];
    asm volatile("s_wait_dscnt 0x0" ::: "memory");

    // ---- coalesced writeback: 2 channels/iter, 16 consecutive w per half-wave ----
    const char* op = (const char*)(out + (size_t)b * CHW + (size_t)h * Wn + w0);
    const int pxr   = lane & 15;
    const int chalf = lane >> 4;
    int oOff = (chalf * HW + pxr) * 4;
#pragma unroll
    for (int i = 0; i < 32; ++i) {
        int c = 2 * i + chalf;
        *(float*)(op + oOff) = buf[c * OSTR + pxr];
        oOff += 2 * HW * 4;
    }
}

extern "C" void kernel_launch(void* const* d_in, const int* in_sizes, int n_in,
                              void* d_out, int out_size, void* d_ws, size_t ws_size,
                              hipStream_t stream) {
    const float* img = (const float*)d_in[0];  // [8,64,256,448]
    const float* flo = (const float*)d_in[1];  // [8,2,256,448]
    float* out = (float*)d_out;                // [8,64,256,448]

    dim3 block(32 * WAVES_PER_BLOCK);
    dim3 grid(NTILES / WAVES_PER_BLOCK);       // 7168 blocks, exact cover
    warp_bilinear_wmma<<<grid, block, 0, stream>>>(img, flo, out);
}